// GMM_70042326663963
// MI455X (gfx1250) — compile-verified
//
#include <hip/hip_runtime.h>
#include <math.h>
#include <stdint.h>

// GMM background-subtraction (K=3, N=8, EM) — one thread per pixel.
// frames: [8,H,W,3] f32, frame: [H,W,3] f32, n_iter: device int scalar.
// out: [2,H,W] f32  (ch0: fg-mask*255, ch1: min distance)

#define TPB 256
#define NFR 8
#define KC  3

__global__ __launch_bounds__(TPB) void gmm_em_kernel(
    const float* __restrict__ frames,
    const float* __restrict__ frame,
    const int*   __restrict__ nIterPtr,
    float* __restrict__ out,
    int HW)
{
    // 8 frame-slices of (256 px * 3 ch) floats = 24 KB staged through LDS.
    __shared__ float smem[NFR * TPB * 3];

    const int tid = threadIdx.x;
    const int p0  = blockIdx.x * TPB;
    const int p   = p0 + tid;

    // Prefetch the classification pixel early; it is consumed only after the
    // EM loop (emits global_prefetch_b8, overlaps with compute).
    if (p < HW) __builtin_prefetch(frame + (size_t)p * 3, 0, 0);

    float x[NFR][3];

    const bool full = (p0 + TPB) <= HW;   // uniform per block
    if (full) {
        // Async global->LDS copy of this block's frame window in 16B chunks.
        // 256 px * 12 B = 3072 B per slice -> 192 chunks; tid<192 is
        // whole-wave uniform (waves 0..5 active, 6..7 idle).
        if (tid < (TPB * 12) / 16) {
            unsigned long long g = (unsigned long long)(uintptr_t)frames
                                 + (unsigned long long)p0 * 12ull
                                 + (unsigned long long)tid * 16ull;
            unsigned l = (unsigned)(uintptr_t)smem + (unsigned)tid * 16u;
            const unsigned long long gstride = (unsigned long long)HW * 12ull;
            #pragma unroll
            for (int n = 0; n < NFR; ++n) {
                asm volatile("global_load_async_to_lds_b128 %0, %1, off"
                             :: "v"(l), "v"(g) : "memory");
                g += gstride;
                l += (unsigned)(TPB * 12);
            }
        }
        asm volatile("s_wait_asynccnt 0x0" ::: "memory");
        __syncthreads();
        #pragma unroll
        for (int n = 0; n < NFR; ++n) {
            const float* s = &smem[n * (TPB * 3) + tid * 3];
            x[n][0] = s[0]; x[n][1] = s[1]; x[n][2] = s[2];
        }
    } else {
        const int pc = (p < HW) ? p : (HW - 1);
        #pragma unroll
        for (int n = 0; n < NFR; ++n) {
            const float* g = frames + ((size_t)n * HW + pc) * 3;
            x[n][0] = g[0]; x[n][1] = g[1]; x[n][2] = g[2];
        }
    }

    const int n_iter = *nIterPtr;                    // uniform scalar load
    const float NORM = 15.749609945722419f;          // sqrt((2*pi)^3)

    // Per-pixel mixture state, fully register resident.
    float w[KC], mk[KC][3], dv[KC];
    #pragma unroll
    for (int k = 0; k < KC; ++k) {
        w[k]  = 1.0f / 3.0f;
        dv[k] = 1.0f;
        const float m0 = (float)(k + 1) * 0.25f;
        mk[k][0] = m0; mk[k][1] = m0; mk[k][2] = m0;
    }

    for (int it = 0; it < n_iter; ++it) {
        // Fold weight/(NORM*dev) and -1/(2*dev^2) into per-k constants.
        float inva[KC], coef[KC];
        #pragma unroll
        for (int k = 0; k < KC; ++k) {
            const float v = dv[k];
            inva[k] = -0.5f / (v * v);
            coef[k] = w[k] / (NORM * v);
        }

        float Rs[KC]    = {0.f, 0.f, 0.f};
        float Sx[KC][3] = {};
        float R[NFR][KC];

        // E-step + weighted sums.
        #pragma unroll
        for (int n = 0; n < NFR; ++n) {
            float rk[KC];
            float rsum = 0.f;
            #pragma unroll
            for (int k = 0; k < KC; ++k) {
                const float d0 = x[n][0] - mk[k][0];
                const float d1 = x[n][1] - mk[k][1];
                const float d2 = x[n][2] - mk[k][2];
                const float q  = d0 * d0 + d1 * d1 + d2 * d2;
                const float r  = coef[k] * __expf(q * inva[k]);  // v_exp_f32
                rk[k] = r;
                rsum += r;
            }
            const float rinv = 1.0f / rsum;
            #pragma unroll
            for (int k = 0; k < KC; ++k) {
                const float r = rk[k] * rinv;
                R[n][k] = r;
                Rs[k]  += r;
                Sx[k][0] += r * x[n][0];
                Sx[k][1] += r * x[n][1];
                Sx[k][2] += r * x[n][2];
            }
        }

        // M-step: new weight/mean, then variance against the NEW mean
        // (reference stores this "variance" back into dev — replicated).
        #pragma unroll
        for (int k = 0; k < KC; ++k) {
            const float invRs = 1.0f / Rs[k];
            w[k] = Rs[k] * (1.0f / (float)NFR);
            mk[k][0] = Sx[k][0] * invRs;
            mk[k][1] = Sx[k][1] * invRs;
            mk[k][2] = Sx[k][2] * invRs;
            float acc = 0.f;
            #pragma unroll
            for (int n = 0; n < NFR; ++n) {
                const float d0 = x[n][0] - mk[k][0];
                const float d1 = x[n][1] - mk[k][1];
                const float d2 = x[n][2] - mk[k][2];
                acc += R[n][k] * (d0 * d0 + d1 * d1 + d2 * d2);
            }
            dv[k] = acc * invRs;
        }
    }

    if (p < HW) {
        const float f0 = frame[(size_t)p * 3 + 0];
        const float f1 = frame[(size_t)p * 3 + 1];
        const float f2 = frame[(size_t)p * 3 + 2];

        float d2k[KC];
        #pragma unroll
        for (int k = 0; k < KC; ++k) {
            const float d0 = f0 - mk[k][0];
            const float d1 = f1 - mk[k][1];
            const float d2 = f2 - mk[k][2];
            d2k[k] = d0 * d0 + d1 * d1 + d2 * d2;
        }
        // argmin (first occurrence on ties); sqrt is monotone so squared
        // distances give the same index.
        int   bi   = 0;
        float best = d2k[0];
        if (d2k[1] < best) { best = d2k[1]; bi = 1; }
        if (d2k[2] < best) { best = d2k[2]; bi = 2; }

        const float r0 = w[0] / dv[0];
        const float r1 = w[1] / dv[1];
        const float r2 = w[2] / dv[2];
        float rmin = fminf(r0, fminf(r1, r2));
        const float rb = (bi == 0) ? r0 : (bi == 1) ? r1 : r2;
        const bool mask2 = (rb <= rmin);

        out[p]      = mask2 ? 255.0f : 0.0f;
        out[HW + p] = sqrtf(best);
    }
}

extern "C" void kernel_launch(void* const* d_in, const int* in_sizes, int n_in,
                              void* d_out, int out_size, void* d_ws, size_t ws_size,
                              hipStream_t stream) {
    const float* frames = (const float*)d_in[0];
    const float* frame  = (const float*)d_in[1];
    const int*   nIter  = (const int*)d_in[2];
    float*       out    = (float*)d_out;

    const int HW = in_sizes[1] / 3;           // 480*640 = 307200
    const int blocks = (HW + TPB - 1) / TPB;  // 1200

    gmm_em_kernel<<<blocks, TPB, 0, stream>>>(frames, frame, nIter, out, HW);
}